// SIR_5652176962199
// MI455X (gfx1250) — compile-verified
//
#include <hip/hip_runtime.h>
#include <hip/hip_bf16.h>

// SIR recurrence: 2047-step serial nonlinear scalar recurrence.
// Latency-bound (dependent-chain) workload -> single wave32, lane 0 runs the
// chain; 4-row register buffering -> 16B-aligned non-temporal b128 stores.
// Division by the constant 1000 is done with the Markstein fma sequence,
// which produces the correctly-rounded IEEE quotient (bit-identical to
// v_div_* lowering) in 3 dependent ops instead of ~10+.

typedef __attribute__((ext_vector_type(4))) float v4f;

__device__ __forceinline__ float div1000_exact(float s0) {
    // r = RN(1/1000); Markstein: correctly-rounded s0/1000 via one fma refine.
    const float r = 1.0f / 1000.0f;      // constant-folded, correctly rounded
    float y0 = s0 * r;
    float e  = fmaf(-1000.0f, y0, s0);   // exact residual (fma required)
    return fmaf(e, r, y0);               // correctly rounded quotient
}

__global__ __launch_bounds__(32) void SIR_5652176962199_kernel(
    const float* __restrict__ x,
    const float* __restrict__ pw2,
    const float* __restrict__ pw3,
    const float* __restrict__ pb3,
    const float* __restrict__ pw4,
    const int*   __restrict__ pn,
    float* __restrict__ out)
{
    if (threadIdx.x != 0) return;

    const float w2 = *pw2;
    const float w3 = *pw3;
    const float b3 = *pb3;
    const float w4 = *pw4;
    const int   n  = *pn;

    float s0 = x[0];
    float s1 = x[1];
    float s2 = x[2];
    float s3 = x[3];
    // slot 4 of the state is written as 0 every step in the reference.

    const int rows = n - 1;          // 2047 output rows
    const int full = rows & ~3;      // groups of 4 rows (80B, 16B-aligned)

    v4f* __restrict__ out4 = reinterpret_cast<v4f*>(out);

    int r = 0;
    for (; r < full; r += 4) {
        float buf[20];
#pragma unroll
        for (int j = 0; j < 4; ++j) {
            // Mirror the reference op order (divide replaced by bit-exact
            // Markstein sequence).
            float a = w2 * (w2 * s0);                                 // lambda2(lambda2(s0))
            float b = w4 * (w4 * s0);                                 // lambda4(lambda4(s0))
            float u = div1000_exact(s0) * s2;                         // s0/1000 * s2
            float c = w3 * (w3 * u + b3) + b3;                        // lambda3(lambda3(...))
            float n0 = s0 - a - b + c;
            float n1 = s1 + a;
            float n2 = s2 - c;
            float n3 = s3 + b;
            s0 = n0; s1 = n1; s2 = n2; s3 = n3;
            buf[j * 5 + 0] = n0;
            buf[j * 5 + 1] = n1;
            buf[j * 5 + 2] = n2;
            buf[j * 5 + 3] = n3;
            buf[j * 5 + 4] = 0.0f;
        }
        // 4 rows = 20 floats = 5 x v4f; base offset r*5 floats is a
        // multiple of 4 floats (16B) because r % 4 == 0.
        const size_t base4 = (size_t)(r * 5) / 4;
#pragma unroll
        for (int q = 0; q < 5; ++q) {
            v4f v;
            v.x = buf[q * 4 + 0];
            v.y = buf[q * 4 + 1];
            v.z = buf[q * 4 + 2];
            v.w = buf[q * 4 + 3];
            // write-once output, never re-read by this kernel: non-temporal
            __builtin_nontemporal_store(v, &out4[base4 + q]);
        }
    }

    // Tail rows (rows % 4 == 3 for n = 2048)
    for (; r < rows; ++r) {
        float a = w2 * (w2 * s0);
        float b = w4 * (w4 * s0);
        float u = div1000_exact(s0) * s2;
        float c = w3 * (w3 * u + b3) + b3;
        float n0 = s0 - a - b + c;
        float n1 = s1 + a;
        float n2 = s2 - c;
        float n3 = s3 + b;
        s0 = n0; s1 = n1; s2 = n2; s3 = n3;
        const size_t base = (size_t)r * 5;
        __builtin_nontemporal_store(n0,  &out[base + 0]);
        __builtin_nontemporal_store(n1,  &out[base + 1]);
        __builtin_nontemporal_store(n2,  &out[base + 2]);
        __builtin_nontemporal_store(n3,  &out[base + 3]);
        __builtin_nontemporal_store(0.0f, &out[base + 4]);
    }
}

extern "C" void kernel_launch(void* const* d_in, const int* in_sizes, int n_in,
                              void* d_out, int out_size, void* d_ws, size_t ws_size,
                              hipStream_t stream) {
    (void)in_sizes; (void)n_in; (void)d_ws; (void)ws_size; (void)out_size;
    const float* x   = (const float*)d_in[0];
    const float* w2  = (const float*)d_in[1];
    const float* w3  = (const float*)d_in[2];
    const float* b3  = (const float*)d_in[3];
    const float* w4  = (const float*)d_in[4];
    const int*   n   = (const int*)  d_in[5];
    float* out = (float*)d_out;

    // Serial recurrence: one wave32, lane 0 owns the dependent chain.
    SIR_5652176962199_kernel<<<1, 32, 0, stream>>>(x, w2, w3, b3, w4, n, out);
}